// GCN_19997367730786
// MI455X (gfx1250) — compile-verified
//
#include <hip/hip_runtime.h>

typedef __attribute__((ext_vector_type(2))) float v2f;
typedef __attribute__((ext_vector_type(8))) float v8f;

#define IN_F 128
#define HID  128
#define NCLS 40
#define SCAN_TPB 256

// ---------------------------------------------------------------- zero fill
__global__ void k_zero_i(int* __restrict__ p, int n) {
    int i = blockIdx.x * blockDim.x + threadIdx.x;
    if (i < n) p[i] = 0;
}

// ---------------------------------------------------------------- degrees
__global__ void k_degrees(const int* __restrict__ src, const int* __restrict__ dst,
                          int* __restrict__ degS, int* __restrict__ degD, int nE) {
    int e = blockIdx.x * blockDim.x + threadIdx.x;
    if (e >= nE) return;
    atomicAdd(&degS[src[e]], 1);
    atomicAdd(&degD[dst[e]], 1);
}

__global__ void k_norms(const int* __restrict__ degS, const int* __restrict__ degD,
                        float* __restrict__ snorm, float* __restrict__ dnorm, int nN) {
    int i = blockIdx.x * blockDim.x + threadIdx.x;
    if (i >= nN) return;
    int ds = degS[i]; if (ds < 1) ds = 1;
    int dd = degD[i]; if (dd < 1) dd = 1;
    snorm[i] = rsqrtf((float)ds);
    dnorm[i] = rsqrtf((float)dd);
}

// ---------------------------------------------------------------- exclusive scan (3-phase)
// phase 1: per-block exclusive scan of degD -> rowStart, block totals -> blockSums
__global__ void k_scan1(const int* __restrict__ deg, int* __restrict__ rowStart,
                        int* __restrict__ blockSums, int nN) {
    __shared__ int sh[SCAN_TPB];
    int t = threadIdx.x;
    int i = blockIdx.x * SCAN_TPB + t;
    int v = (i < nN) ? deg[i] : 0;
    sh[t] = v;
    __syncthreads();
#pragma unroll
    for (int off = 1; off < SCAN_TPB; off <<= 1) {
        int add = (t >= off) ? sh[t - off] : 0;
        __syncthreads();
        sh[t] += add;
        __syncthreads();
    }
    if (i < nN) rowStart[i] = sh[t] - v;            // exclusive within block
    if (t == SCAN_TPB - 1) blockSums[blockIdx.x] = sh[t];
}

// phase 2: single-block exclusive scan of blockSums in place (nB <= 1024)
__global__ void k_scan2(int* __restrict__ blockSums, int nB) {
    __shared__ int sh[1024];
    int t = threadIdx.x;
    int v = (t < nB) ? blockSums[t] : 0;
    sh[t] = v;
    __syncthreads();
    for (int off = 1; off < 1024; off <<= 1) {
        int add = (t >= off) ? sh[t - off] : 0;
        __syncthreads();
        sh[t] += add;
        __syncthreads();
    }
    if (t < nB) blockSums[t] = sh[t] - v;           // exclusive
}

// phase 3: add block offsets; append total
__global__ void k_scan3(int* __restrict__ rowStart, const int* __restrict__ blockSums,
                        int nN, int nE) {
    int i = blockIdx.x * blockDim.x + threadIdx.x;
    if (i < nN) rowStart[i] += blockSums[i / SCAN_TPB];
    if (i == 0) rowStart[nN] = nE;
}

// ---------------------------------------------------------------- CSR bucket fill
__global__ void k_bucket(const int* __restrict__ src, const int* __restrict__ dst,
                         const int* __restrict__ rowStart, int* __restrict__ fill,
                         int* __restrict__ edgeSrc, int nE) {
    int e = blockIdx.x * blockDim.x + threadIdx.x;
    if (e >= nE) return;
    int d = dst[e];
    int p = rowStart[d] + atomicAdd(&fill[d], 1);
    edgeSrc[p] = src[e];
}

// ---------------------------------------------------------------- aggregation (atomic-free)
// One wave per destination node. Lane l accumulates features [4l, 4l+4) in
// registers over all in-edges; one float4 store. agg[i] = sum_e snorm[src]*feat[src].
__global__ void k_aggregate(const float* __restrict__ feat, const float* __restrict__ snorm,
                            const int* __restrict__ edgeSrc, const int* __restrict__ rowStart,
                            float* __restrict__ agg, int nN) {
    int node = blockIdx.x * (blockDim.x >> 5) + (threadIdx.x >> 5);
    int lane = threadIdx.x & 31;
    if (node >= nN) return;
    int beg = rowStart[node];
    int end = rowStart[node + 1];
    float4 acc = make_float4(0.f, 0.f, 0.f, 0.f);
    for (int j = beg; j < end; ++j) {
        int s = edgeSrc[j];
        if (j + 1 < end) {      // prefetch next source row (global_prefetch_b8)
            __builtin_prefetch(feat + (size_t)edgeSrc[j + 1] * IN_F + lane * 4, 0, 0);
        }
        float sn = snorm[s];
        const float4 v = *reinterpret_cast<const float4*>(feat + (size_t)s * IN_F + lane * 4);
        acc.x += v.x * sn;
        acc.y += v.y * sn;
        acc.z += v.z * sn;
        acc.w += v.w * sn;
    }
    *reinterpret_cast<float4*>(agg + (size_t)node * IN_F + lane * 4) = acc;
}

// ---------------------------------------------------------------- layer-1 GEMM
// One wave per 16-node tile. D(16x128) = (agg*dnorm)(16x128) @ W(128x128) + b, ReLU.
// Exact fp32 matmul via V_WMMA_F32_16X16X4_F32, K swept in steps of 4.
__global__ void k_gemm128_relu(const float* __restrict__ agg, const float* __restrict__ dnorm,
                               const float* __restrict__ W, const float* __restrict__ bias,
                               float* __restrict__ out, int nTiles) {
    int tile = blockIdx.x * (blockDim.x >> 5) + (threadIdx.x >> 5);
    if (tile >= nTiles) return;
    int lane = threadIdx.x & 31;
    int l    = lane & 15;
    int half = lane >> 4;

    int arow = tile * 16 + l;               // A-matrix row owned by this lane
    float dn = dnorm[arow];
    const float* ap = agg + (size_t)arow * HID;

    v8f acc[8] = {};                        // 8 N-tiles of 16 -> full 128 cols

#pragma unroll 4
    for (int kk = 0; kk < HID / 4; ++kk) {
        int kb = kk * 4 + half * 2;         // K-pair base for this lane half
        v2f a;
        a.x = ap[kb]     * dn;
        a.y = ap[kb + 1] * dn;
        const float* w0 = W + (size_t)kb * HID;
#pragma unroll
        for (int n = 0; n < 8; ++n) {
            v2f b;
            b.x = w0[n * 16 + l];
            b.y = w0[HID + n * 16 + l];
            acc[n] = __builtin_amdgcn_wmma_f32_16x16x4_f32(
                false, a, false, b, (short)0, acc[n], false, false);
        }
    }

#pragma unroll
    for (int n = 0; n < 8; ++n) {
        int col = n * 16 + l;
        float bv = bias[col];
#pragma unroll
        for (int r = 0; r < 8; ++r) {
            int orow = tile * 16 + half * 8 + r;   // D layout: M = r + 8*half
            float v = acc[n][r] + bv;
            out[(size_t)orow * HID + col] = fmaxf(v, 0.f);
        }
    }
}

// ---------------------------------------------------------------- layer-2 GEMM
// D(16x40) = (agg*dnorm)(16x128) @ W2(128x40) + b2. 3 N-tiles, cols >= 40 are
// clamped on load (keeps EXEC all-1s at the WMMA) and masked on store.
__global__ void k_gemm40(const float* __restrict__ agg, const float* __restrict__ dnorm,
                         const float* __restrict__ W, const float* __restrict__ bias,
                         float* __restrict__ out, int nTiles) {
    int tile = blockIdx.x * (blockDim.x >> 5) + (threadIdx.x >> 5);
    if (tile >= nTiles) return;
    int lane = threadIdx.x & 31;
    int l    = lane & 15;
    int half = lane >> 4;

    int arow = tile * 16 + l;
    float dn = dnorm[arow];
    const float* ap = agg + (size_t)arow * HID;

    v8f acc[3] = {};

#pragma unroll 4
    for (int kk = 0; kk < HID / 4; ++kk) {
        int kb = kk * 4 + half * 2;
        v2f a;
        a.x = ap[kb]     * dn;
        a.y = ap[kb + 1] * dn;
        const float* w0 = W + (size_t)kb * NCLS;
#pragma unroll
        for (int n = 0; n < 3; ++n) {
            int c  = n * 16 + l;
            int cc = c < NCLS ? c : (NCLS - 1);   // clamp, no EXEC divergence
            v2f b;
            b.x = w0[cc];
            b.y = w0[NCLS + cc];
            acc[n] = __builtin_amdgcn_wmma_f32_16x16x4_f32(
                false, a, false, b, (short)0, acc[n], false, false);
        }
    }

#pragma unroll
    for (int n = 0; n < 3; ++n) {
        int col = n * 16 + l;
        if (col < NCLS) {
            float bv = bias[col];
#pragma unroll
            for (int r = 0; r < 8; ++r) {
                int orow = tile * 16 + half * 8 + r;
                out[(size_t)orow * NCLS + col] = acc[n][r] + bv;
            }
        }
    }
}

// ---------------------------------------------------------------- launcher
extern "C" void kernel_launch(void* const* d_in, const int* in_sizes, int n_in,
                              void* d_out, int out_size, void* d_ws, size_t ws_size,
                              hipStream_t stream) {
    const float* x   = (const float*)d_in[0];
    const float* W1  = (const float*)d_in[1];
    const float* b1  = (const float*)d_in[2];
    const float* W2  = (const float*)d_in[3];
    const float* b2  = (const float*)d_in[4];
    const int*   src = (const int*)d_in[5];
    const int*   dst = (const int*)d_in[6];
    float*       out = (float*)d_out;

    const int nN = in_sizes[0] / IN_F;      // 100000
    const int nE = in_sizes[5];             // 1600000
    const int nB = (nN + SCAN_TPB - 1) / SCAN_TPB;   // 391 scan blocks (<=1024)

    // ---------------- workspace layout (16B-aligned slices) ----------------
    char* ws = (char*)d_ws;
    const size_t featBytes = (size_t)nN * IN_F * sizeof(float);   // 51.2 MB
    size_t off = 0;
    float* agg      = (float*)(ws + off); off += featBytes;
    float* h        = (float*)(ws + off); off += featBytes;
    float* snorm    = (float*)(ws + off); off += (size_t)nN * 4;
    float* dnorm    = (float*)(ws + off); off += (size_t)nN * 4;
    int*   degS     = (int*)  (ws + off); off += (size_t)nN * 4;
    int*   degD     = (int*)  (ws + off); off += (size_t)nN * 4;
    int*   fill     = (int*)  (ws + off); off += (size_t)nN * 4;
    int*   rowStart = (int*)  (ws + off); off += ((size_t)nN + 4) * 4;  // nN+1, padded
    int*   edgeSrc  = (int*)  (ws + off); off += (size_t)nE * 4;
    int*   blockSums= (int*)  (ws + off); off += 1024 * 4;

    const int TPB = 256;

    // degrees + norms (degS,degD,fill are contiguous -> one zero pass)
    k_zero_i<<<(3 * nN + TPB - 1) / TPB, TPB, 0, stream>>>(degS, 3 * nN);
    k_degrees<<<(nE + TPB - 1) / TPB, TPB, 0, stream>>>(src, dst, degS, degD, nE);
    k_norms<<<(nN + TPB - 1) / TPB, TPB, 0, stream>>>(degS, degD, snorm, dnorm, nN);

    // build dst-CSR once (shared by both layers)
    k_scan1<<<nB, SCAN_TPB, 0, stream>>>(degD, rowStart, blockSums, nN);
    k_scan2<<<1, 1024, 0, stream>>>(blockSums, nB);
    k_scan3<<<(nN + TPB - 1) / TPB, TPB, 0, stream>>>(rowStart, blockSums, nN, nE);
    k_bucket<<<(nE + TPB - 1) / TPB, TPB, 0, stream>>>(src, dst, rowStart, fill, edgeSrc, nE);

    const int aggBlks  = (nN + (TPB / 32) - 1) / (TPB / 32);   // wave per node
    const int nTiles   = nN / 16;                              // 6250
    const int gemmBlks = (nTiles + (TPB / 32) - 1) / (TPB / 32);

    // layer 1: aggregate -> WMMA GEMM + bias + ReLU
    k_aggregate<<<aggBlks, TPB, 0, stream>>>(x, snorm, edgeSrc, rowStart, agg, nN);
    k_gemm128_relu<<<gemmBlks, TPB, 0, stream>>>(agg, dnorm, W1, b1, h, nTiles);

    // layer 2: aggregate -> WMMA GEMM + bias
    k_aggregate<<<aggBlks, TPB, 0, stream>>>(h, snorm, edgeSrc, rowStart, agg, nN);
    k_gemm40<<<gemmBlks, TPB, 0, stream>>>(agg, dnorm, W2, b2, out, nTiles);
}